// DifferentiableGaborKernel_91027536871515
// MI455X (gfx1250) — compile-verified
//
#include <hip/hip_runtime.h>
#include <stdint.h>

typedef float v2f __attribute__((ext_vector_type(2)));
typedef float v8f __attribute__((ext_vector_type(8)));

#define HH    320
#define WW    320
#define KS    31
#define HALF  15
#define HOUT  289
#define WOUT  289
#define NTAP  961            // 31*31
#define SEGS  3              // 289 pixels / 128 per block -> 3 segments per row
#define SEGW  128            // 8 waves * 16 pixels
#define TILEW 160            // SEGW + 30 halo, rounded to 32 (LDS row stride)
#define THRESH_V 55.0f

// ---- order-preserving float <-> uint key (for atomic min/max) --------------
__device__ __forceinline__ unsigned fkey(float f) {
    unsigned u = __float_as_uint(f);
    return (u & 0x80000000u) ? ~u : (u | 0x80000000u);
}
__device__ __forceinline__ float unkey(unsigned k) {
    unsigned u = (k & 0x80000000u) ? (k & 0x7FFFFFFFu) : ~k;
    return __uint_as_float(u);
}

// ---- one Gabor tap + matching LDS patch value for linear tap index t -------
__device__ __forceinline__ void gabor_tap_patch(
    int t, int colbase_l, float ct, float st, float f0,
    const float* __restrict__ tile, float& tap, float& patch)
{
    int tc = min(t, NTAP - 1);          // clamp for K padding (961 -> 964)
    int ry = tc / KS;                   // kernel row (yy axis)
    int rx = tc - ry * KS;              // kernel col (xx axis)
    float xv = (float)(rx - HALF);
    float yv = (float)(ry - HALF);
    float x_t = xv * ct + yv * st;
    float y_t = yv * ct - xv * st;
    float fl  = f0 * (1.0f + y_t * (1.0f / 45.0f));        // f0 + f0/3 * y_t/15
    float gy  = (1.0f + 0.04f * fabsf(y_t)) * y_t;         // gamma * y_t
    float env = __expf((x_t * x_t + gy * gy) * (-1.0f / 72.0f));
    float tp  = env * __cosf(6.2831853071795864f * fl * x_t);
    tap = (t < NTAP) ? tp : 0.0f;       // zero padded-K taps
    patch = tile[ry * TILEW + colbase_l + rx];             // ds_load from LDS
}

// ---- stage 1: per-pixel Gabor responses via V_WMMA_F32_16X16X4_F32 ---------
// Block = 8 waves = 128 consecutive output pixels of one row. The 31x160
// fprint footprint is staged once into LDS; each wave then runs the K-loop:
// A(16x4) = patches (from LDS), B(4x16) = per-pixel taps, diag(A*B) = result.
__global__ void __launch_bounds__(256)
gabor_wmma_kernel(const float* __restrict__ fprint,
                  const int*   __restrict__ freq_map,
                  const int*   __restrict__ theta_map,
                  float*       __restrict__ out /*interior store*/)
{
    __shared__ float tile[KS * TILEW];           // 31*160*4 = 19.4 KB

    const int row = blockIdx.x / SEGS;           // patch top row == output row idx
    const int seg = blockIdx.x - row * SEGS;
    const int pj0 = seg * SEGW;                  // first output pixel of block

    // ---- cooperative LDS staging of the fprint footprint -------------------
    // rows [row, row+31), cols [pj0, pj0+160). All addresses stay inside the
    // 320x320 image (max flat index 318*320+415 = 102175 < 102400); columns
    // past 319 wrap into the next image row but only ever feed diagonal
    // entries of invalid (not-stored) pixels.
    {
        const int base = row * WW + pj0;
        for (int i = threadIdx.x; i < KS * TILEW; i += 256) {
            const int r = i / TILEW;
            const int c = i - r * TILEW;
            tile[i] = fprint[base + r * WW + c];
        }
    }
    __syncthreads();

    const int lane = threadIdx.x & 31;
    const int wib  = threadIdx.x >> 5;           // wave-in-block: group 0..7
    const int m    = lane & 15;                  // pixel index in group
    const int kb   = (lane >> 4) << 1;           // K offset inside 4-chunk: 0 or 2
    const int colbase_l = wib * 16 + m;          // LDS column base for this lane

    // per-pixel parameters (pixel n == m): theta/freq at output coords
    const int pcol = min(pj0 + wib * 16 + m + HALF, WW - 1);
    const int prow = row + HALF;
    const float theta = (float)theta_map[prow * WW + pcol] *
                        (3.14159265358979323846f / 180.0f);
    const float f0 = 0.025f + 0.0015f * (float)freq_map[prow * WW + pcol];
    const float ct = __cosf(theta);
    const float st = __sinf(theta);

    v8f c = {0.f, 0.f, 0.f, 0.f, 0.f, 0.f, 0.f, 0.f};

    for (int t0 = 0; t0 < NTAP; t0 += 4) {
        float ta_tap, ta_pat, tb_tap, tb_pat;
        gabor_tap_patch(t0 + kb,     colbase_l, ct, st, f0, tile, ta_tap, ta_pat);
        gabor_tap_patch(t0 + kb + 1, colbase_l, ct, st, f0, tile, tb_tap, tb_pat);
        v2f a = { ta_pat, tb_pat };   // A: M=m, K=kb,kb+1
        v2f b = { ta_tap, tb_tap };   // B: N=m, K=kb,kb+1
        c = __builtin_amdgcn_wmma_f32_16x16x4_f32(
                /*neg_a=*/false, a, /*neg_b=*/false, b,
                /*c_mod=*/(short)0, c, /*reuse_a=*/false, /*reuse_b=*/false);
    }

    // diagonal extraction: diag(v) @ lane v in c[v] (v<8); diag(v+8) @ lane v+24 in c[v]
    float r = 0.0f;
#pragma unroll
    for (int v = 0; v < 8; ++v) {
        if (lane == v)      r = c[v];
        if (lane == v + 24) r = c[v];
    }
    const int  pix  = (lane < 8) ? lane : (lane - 16);
    const bool want = (lane < 8) || (lane >= 24);
    const int  pj   = pj0 + wib * 16 + pix;
    if (want && pj < WOUT)
        out[(row + HALF) * WW + (pj + HALF)] = r;   // store into final image spot
}

// ---- stage 0: reset min/max key slots (graph-replay deterministic) ---------
__global__ void init_keys_kernel(unsigned* __restrict__ keys) {
    keys[0] = 0xFFFFFFFFu;   // running-min key
    keys[1] = 0x00000000u;   // running-max key
}

// ---- stage 2: global min/max over composite image --------------------------
__global__ void minmax_kernel(const float* __restrict__ fprint,
                              const float* __restrict__ interior,
                              unsigned*    __restrict__ keys)
{
    const int idx = blockIdx.x * blockDim.x + threadIdx.x;
    unsigned kmin = 0xFFFFFFFFu, kmax = 0u;
    if (idx < HH * WW) {
        const int y = idx / WW;
        const int x = idx - y * WW;
        const bool in = (y >= HALF) & (y < HALF + HOUT) & (x >= HALF) & (x < HALF + WOUT);
        const float v = in ? interior[idx] : fprint[idx];
        const unsigned k = fkey(v);
        kmin = k; kmax = k;
    }
#pragma unroll
    for (int off = 16; off > 0; off >>= 1) {
        kmin = min(kmin, (unsigned)__shfl_xor((int)kmin, off, 32));
        kmax = max(kmax, (unsigned)__shfl_xor((int)kmax, off, 32));
    }
    if ((threadIdx.x & 31) == 0) {
        atomicMin(&keys[0], kmin);
        atomicMax(&keys[1], kmax);
    }
}

// ---- stage 3: normalize + threshold -----------------------------------------
__global__ void finalize_kernel(const float*    __restrict__ fprint,
                                const unsigned* __restrict__ keys,
                                float*          __restrict__ out)
{
    const int idx = blockIdx.x * blockDim.x + threadIdx.x;
    if (idx >= HH * WW) return;
    const float mn = unkey(keys[0]);
    const float mx = unkey(keys[1]);
    const int y = idx / WW;
    const int x = idx - y * WW;
    const bool in = (y >= HALF) & (y < HALF + HOUT) & (x >= HALF) & (x < HALF + WOUT);
    float v = (in ? out[idx] : fprint[idx]) - mn;   // out = out - min
    const float range = mx - mn;                    // == max(out - min)
    if (range != 0.0f) v = v / range * 100.0f;
    out[idx] = (v > THRESH_V) ? 100.0f : 0.0f;
}

extern "C" void kernel_launch(void* const* d_in, const int* in_sizes, int n_in,
                              void* d_out, int out_size, void* d_ws, size_t ws_size,
                              hipStream_t stream)
{
    const float* fprint    = (const float*)d_in[0];
    const int*   freq_map  = (const int*)  d_in[1];
    const int*   theta_map = (const int*)  d_in[2];
    float*       out       = (float*)d_out;
    unsigned*    keys      = (unsigned*)d_ws;       // 8 bytes of scratch

    init_keys_kernel<<<1, 1, 0, stream>>>(keys);

    const int blocks = HOUT * SEGS;                 // 289 rows * 3 segments = 867
    gabor_wmma_kernel<<<blocks, 256, 0, stream>>>(fprint, freq_map, theta_map, out);

    const int n = HH * WW;
    minmax_kernel<<<(n + 255) / 256, 256, 0, stream>>>(fprint, out, keys);
    finalize_kernel<<<(n + 255) / 256, 256, 0, stream>>>(fprint, keys, out);
}